// MultiHeadAttentionLayerGritSparse_1675037246120
// MI455X (gfx1250) — compile-verified
//
#include <hip/hip_runtime.h>
#include <hip/hip_bf16.h>
#include <math.h>

#define N_NODES 50000
#define N_EDGES 800000
#define IN_DIM  64
#define NH      8
#define ND      8
#define HD      64      // NH*ND
#define CLAMPV  5.0f
#define EPSV    1e-16f

#define EDGE_TILE   128                 // edges per tile
#define TILES_TOTAL (N_EDGES / EDGE_TILE)   // 6250
#define TILES_PER_WG 5
#define EDGE_WGS    (TILES_TOTAL / TILES_PER_WG)  // 1250

typedef __attribute__((ext_vector_type(2))) float    v2f;
typedef __attribute__((ext_vector_type(8))) float    v8f;
typedef __attribute__((ext_vector_type(4))) unsigned v4u;
typedef __attribute__((ext_vector_type(8))) int      v8i;
typedef __attribute__((ext_vector_type(4))) int      v4i;

// ---- CDNA5 fp32 WMMA: D(16x16,f32) = A(16x4,f32) * B(4x16,f32) + C ----
__device__ __forceinline__ v8f wmma_f32(v2f a, v2f b, v8f c) {
  return __builtin_amdgcn_wmma_f32_16x16x4_f32(
      /*neg_a=*/false, a, /*neg_b=*/false, b,
      /*c_mod=*/(short)0, c, /*reuse_a=*/false, /*reuse_b=*/false);
}

// ---- CDNA5 async global->LDS copy (ASYNCcnt-tracked), GV addressing ----
__device__ __forceinline__ void async_ld_b128(unsigned lds_off,
                                              unsigned long long gaddr) {
  asm volatile("global_load_async_to_lds_b128 %0, %1, off"
               :: "v"(lds_off), "v"(gaddr)
               : "memory");
}
__device__ __forceinline__ void wait_async_le8() {
  asm volatile("s_wait_asynccnt 0x8" ::: "memory");
}
__device__ __forceinline__ void wait_async_0() {
  asm volatile("s_wait_asynccnt 0x0" ::: "memory");
}

// ---- CDNA5 Tensor Data Mover: 2D global->LDS tile DMA (TENSORcnt) ----
// clang-23 toolchain: 6-arg form (g0, g1, g2, g3, g4, cpol)
#if __has_builtin(__builtin_amdgcn_tensor_load_to_lds)
#define HAVE_TDM 1
__device__ __forceinline__ void tdm_load_2d_f32(unsigned lds_addr,
                                                const void* gptr,
                                                unsigned dim0, unsigned dim1) {
  const unsigned long long ga = (unsigned long long)gptr;
  v4u g0;
  g0.x = 1u;                                           // count=1 (valid user D#)
  g0.y = lds_addr;                                     // LDS byte address
  g0.z = (unsigned)(ga & 0xFFFFFFFFull);               // global_addr[31:0]
  g0.w = (unsigned)((ga >> 32) & 0x01FFFFFFull)        // global_addr[56:32]
         | 0x80000000u;                                // type=2 ("image")
  v8i g1;
  g1[0] = (int)(2u << 16);                             // data_size=4B, mask=0
  g1[1] = (int)((dim0 & 0xFFFFu) << 16);               // tensor_dim0[15:0]
  g1[2] = (int)((dim0 >> 16) | ((dim1 & 0xFFFFu) << 16)); // dim0 hi | dim1 lo
  g1[3] = (int)((dim1 >> 16) | ((dim0 & 0xFFFFu) << 16)); // dim1 hi | tile_dim0
  g1[4] = (int)(dim1 & 0xFFFFu);                       // tile_dim1 (tile_dim2=0)
  g1[5] = (int)dim0;                                   // tensor_dim0_stride lo
  g1[6] = 0;                                           // stride0 hi | stride1 lo
  g1[7] = 0;
  const v4i z4 = {0, 0, 0, 0};                         // groups 2/3 unused (2D)
  const v8i z8 = {0, 0, 0, 0, 0, 0, 0, 0};
  __builtin_amdgcn_tensor_load_to_lds(g0, g1, z4, z4, z8, 0);
}
#endif

// order-preserving float <-> uint mapping for atomic segment-max
__device__ __forceinline__ unsigned ford(float f) {
  unsigned u = __float_as_uint(f);
  return (u & 0x80000000u) ? ~u : (u | 0x80000000u);
}
__device__ __forceinline__ float forddec(unsigned u) {
  return (u & 0x80000000u) ? __uint_as_float(u ^ 0x80000000u)
                           : __uint_as_float(~u);
}

// ---------------- init: zero accumulators, seed segment-max ----------------
__global__ __launch_bounds__(256) void k_init(float* __restrict__ wV,
                                              float* __restrict__ ssum,
                                              float* __restrict__ rowV,
                                              unsigned* __restrict__ smax) {
  const int i = blockIdx.x * blockDim.x + threadIdx.x;
  if (i < N_NODES * HD) { wV[i] = 0.0f; rowV[i] = 0.0f; }
  if (i < N_NODES * NH) { ssum[i] = 0.0f; smax[i] = ford(-CLAMPV); }
}

// ---------------- fused Q/K/V projection (x @ W + b) ----------------------
// one wave per (16-node tile, 16-col tile of [Wq|Wk|Wv])
__global__ __launch_bounds__(32) void k_qkv(
    const float* __restrict__ x,
    const float* __restrict__ Wq, const float* __restrict__ bq,
    const float* __restrict__ Wk, const float* __restrict__ bk,
    const float* __restrict__ Wv, const float* __restrict__ bv,
    float* __restrict__ Qh, float* __restrict__ Kh, float* __restrict__ Vh) {
  const int lane = threadIdx.x & 31;
  const int half = lane >> 4;     // 0: lanes 0-15, 1: lanes 16-31
  const int r    = lane & 15;
  const int bm   = blockIdx.x;    // node tile (50000/16 = 3125 exact)
  const int bt   = blockIdx.y;    // 0..11 across [Wq|Wk|Wv] columns

  const float* W; const float* bias; float* Out;
  if (bt < 4)      { W = Wq; bias = bq; Out = Qh; }
  else if (bt < 8) { W = Wk; bias = bk; Out = Kh; }
  else             { W = Wv; bias = bv; Out = Vh; }
  const int cb = (bt & 3) * 16;

  v8f acc = {};
  const float* xrow = x + (bm * 16 + r) * IN_DIM;
#pragma unroll
  for (int k = 0; k < 16; ++k) {
    const int k0 = k * 4 + 2 * half;            // A: VGPRv holds K = v + 2*half
    v2f a = *(const v2f*)(xrow + k0);
    v2f b;
    b.x = W[(k0 + 0) * HD + cb + r];            // B mirrored layout
    b.y = W[(k0 + 1) * HD + cb + r];
    acc = wmma_f32(a, b, acc);
  }
  const float bb = bias[cb + r];
#pragma unroll
  for (int v = 0; v < 8; ++v)                   // C/D: M = v + 8*half, N = r
    Out[(bm * 16 + v + 8 * half) * HD + cb + r] = acc[v] + bb;
}

// ------- edge GEMM (edge_attr @ We + be) fused with score epilogue --------
// Persistent WGs: each processes TILES_PER_WG tiles of 128 edges with
// double-buffered async global->LDS staging of the A (edge_attr) tiles.
// We (64x128) is DMA'd into LDS by the Tensor Data Mover.
__global__ __launch_bounds__(256) void k_edge(
    const float* __restrict__ edge_attr,
    const float* __restrict__ We, const float* __restrict__ be,
    const int* __restrict__ src, const int* __restrict__ dst,
    const float* __restrict__ Qh, const float* __restrict__ Kh,
    const float* __restrict__ Aw,
    float* __restrict__ wE, float* __restrict__ sclip,
    unsigned* __restrict__ smax) {
  __shared__ float sWe[IN_DIM * 2 * HD];            // 64 x 128 = 32KB
  __shared__ float sbe[2 * HD];
  __shared__ float sA[2][EDGE_TILE * IN_DIM];       // 2 x 32KB double buffer

#if defined(HAVE_TDM)
  if (threadIdx.x < 32) {                           // one TDM DMA per WG
    tdm_load_2d_f32((unsigned)(size_t)(&sWe[0]), We,
                    /*dim0=*/2 * HD, /*dim1=*/IN_DIM);
    __builtin_amdgcn_s_wait_tensorcnt(0);
  }
#else
  for (int i = threadIdx.x; i < IN_DIM * 2 * HD; i += 256) sWe[i] = We[i];
#endif
  if (threadIdx.x < 2 * HD) sbe[threadIdx.x] = be[threadIdx.x];

  const int lane = threadIdx.x & 31;
  const int wave = threadIdx.x >> 5;
  const int half = lane >> 4;
  const int n    = lane & 15;
  const int d    = n & 7;

  const int tile0 = blockIdx.x * TILES_PER_WG;

  // issue async copy of tile (tile0+0) into buffer 0
  {
    const unsigned long long g0 =
        (unsigned long long)(edge_attr + (size_t)tile0 * EDGE_TILE * IN_DIM);
    const unsigned l0 = (unsigned)(size_t)(&sA[0][0]);
#pragma unroll
    for (int j = 0; j < 8; ++j) {
      const unsigned c = (unsigned)threadIdx.x + (unsigned)j * 256u;  // 16B chunks
      async_ld_b128(l0 + c * 16u, g0 + (unsigned long long)c * 16ull);
    }
  }

  for (int t = 0; t < TILES_PER_WG; ++t) {
    const int tile = tile0 + t;
    const int buf  = t & 1;

    // prefetch next tile into the other buffer, then retire current tile's 8
    if (t + 1 < TILES_PER_WG) {
      const unsigned long long gn =
          (unsigned long long)(edge_attr +
                               (size_t)(tile + 1) * EDGE_TILE * IN_DIM);
      const unsigned ln = (unsigned)(size_t)(&sA[buf ^ 1][0]);
#pragma unroll
      for (int j = 0; j < 8; ++j) {
        const unsigned c = (unsigned)threadIdx.x + (unsigned)j * 256u;
        async_ld_b128(ln + c * 16u, gn + (unsigned long long)c * 16ull);
      }
      wait_async_le8();        // oldest 8 (current tile) complete, in order
    } else {
      wait_async_0();
    }
    __syncthreads();           // all waves' staged data (TDM + async) visible

    const int eb = tile * EDGE_TILE + wave * 16;

    v8f acc[8];
#pragma unroll
    for (int q = 0; q < 8; ++q) acc[q] = (v8f){};

    const float* aBase = &sA[buf][(wave * 16 + n) * IN_DIM];
#pragma unroll 4
    for (int k = 0; k < 16; ++k) {
      const int k0 = k * 4 + 2 * half;
      v2f a = *(const v2f*)(aBase + k0);           // ds_load_b64
#pragma unroll
      for (int q = 0; q < 8; ++q) {
        v2f b;
        b.x = sWe[(k0 + 0) * 128 + q * 16 + n];
        b.y = sWe[(k0 + 1) * 128 + q * 16 + n];
        acc[q] = wmma_f32(a, b, acc[q]);
      }
    }

    // per-edge endpoints for the 8 C-rows this half owns
    int esrc[8], edst[8];
#pragma unroll
    for (int v = 0; v < 8; ++v) {
      const int e = eb + v + 8 * half;
      esrc[v] = src[e];
      edst[v] = dst[e];
    }

    // epilogue: col (h*16+n): n<8 -> E_w[h,n], n>=8 -> E_b[h,n-8]
#pragma unroll
    for (int h = 0; h < 8; ++h) {
      const float bias = sbe[h * 16 + n];
      const float awh  = Aw[d * NH + h];
#pragma unroll
      for (int v = 0; v < 8; ++v) {
        const int e     = eb + v + 8 * half;
        const float val = acc[h][v] + bias;
        const float ebv = __shfl_xor(val, 8, 32);  // lanes n<8 get E_b[h,d]
        float partial = 0.0f;
        if (n < 8) {
          const int hd = h * ND + d;
          const float kq = Kh[esrc[v] * HD + hd] + Qh[edst[v] * HD + hd];
          float sc = kq * val;                     // (K+Q)*E_w
          sc = copysignf(sqrtf(fabsf(sc)), sc);    // signed sqrt
          const float et = sc + ebv;               // e_t
          wE[e * HD + hd] = et;                    // wE output == e_t
          partial = et * awh;                      // einsum with Aw
        }
        partial += __shfl_xor(partial, 1, 32);
        partial += __shfl_xor(partial, 2, 32);
        partial += __shfl_xor(partial, 4, 32);
        if (n == 0) {
          const float s = fminf(fmaxf(partial, -CLAMPV), CLAMPV);
          sclip[e * NH + h] = s;
          atomicMax(&smax[edst[v] * NH + h], ford(s));
        }
      }
    }
    __syncthreads();   // everyone done reading sA[buf] before it is refilled
  }
}

// ---------------- softmax denominator ----------------
__global__ __launch_bounds__(256) void k_ssum(
    const int* __restrict__ dst, const float* __restrict__ sclip,
    const unsigned* __restrict__ smax, float* __restrict__ ssum) {
  const int i = blockIdx.x * blockDim.x + threadIdx.x;
  if (i >= N_EDGES * NH) return;
  const int e = i >> 3, h = i & 7;
  const int nd = dst[e];
  const float m = forddec(smax[nd * NH + h]);
  atomicAdd(&ssum[nd * NH + h], __expf(sclip[i] - m));
}

// ---------------- attention scatter (msg + rowV) ----------------
__global__ __launch_bounds__(256) void k_scatter(
    const int* __restrict__ src, const int* __restrict__ dst,
    const float* __restrict__ sclip, const unsigned* __restrict__ smax,
    const float* __restrict__ ssum, const float* __restrict__ Vh,
    const float* __restrict__ wE, float* __restrict__ wV,
    float* __restrict__ rowV) {
  const int i = blockIdx.x * blockDim.x + threadIdx.x;
  if (i >= N_EDGES * NH) return;
  const int e = i >> 3, h = i & 7;
  const int ns = src[e], nd = dst[e];
  const float m = forddec(smax[nd * NH + h]);
  const float p = __expf(sclip[i] - m);
  const float attn = p / (ssum[nd * NH + h] + EPSV);
#pragma unroll
  for (int d = 0; d < ND; ++d) {
    const int hd = h * ND + d;
    atomicAdd(&wV[nd * HD + hd], Vh[ns * HD + hd] * attn);
    atomicAdd(&rowV[nd * HD + hd], wE[e * HD + hd] * attn);
  }
}

// ---------------- final per-node einsum: wV += rowV @ VeRow ----------------
__global__ __launch_bounds__(256) void k_final(
    const float* __restrict__ rowV, const float* __restrict__ VeRow,
    float* __restrict__ wV) {
  const int i = blockIdx.x * blockDim.x + threadIdx.x;
  if (i >= N_NODES * HD) return;
  const int nn = i >> 6, h = (i >> 3) & 7, c = i & 7;
  float acc = 0.0f;
#pragma unroll
  for (int d = 0; d < ND; ++d)
    acc += rowV[nn * HD + h * ND + d] * VeRow[d * HD + h * ND + c];
  wV[i] += acc;
}

extern "C" void kernel_launch(void* const* d_in, const int* in_sizes, int n_in,
                              void* d_out, int out_size, void* d_ws, size_t ws_size,
                              hipStream_t stream) {
  const float* x         = (const float*)d_in[0];
  const float* edge_attr = (const float*)d_in[1];
  const int*   eidx      = (const int*)  d_in[2];
  const float* Wq = (const float*)d_in[3];  const float* bq = (const float*)d_in[4];
  const float* Wk = (const float*)d_in[5];  const float* bk = (const float*)d_in[6];
  const float* Wv = (const float*)d_in[7];  const float* bv = (const float*)d_in[8];
  const float* We = (const float*)d_in[9];  const float* be = (const float*)d_in[10];
  const float* Aw = (const float*)d_in[11];
  const float* VeRow = (const float*)d_in[12];
  const int* src = eidx;
  const int* dst = eidx + N_EDGES;

  float* wV = (float*)d_out;                        // [N, 64]
  float* wE = (float*)d_out + N_NODES * HD;         // [E, 64]

  // workspace layout (floats): 20M total = 80 MB
  float* Qh    = (float*)d_ws;
  float* Kh    = Qh + (size_t)N_NODES * HD;
  float* Vh    = Kh + (size_t)N_NODES * HD;
  float* sclip = Vh + (size_t)N_NODES * HD;         // [E, 8]
  unsigned* smax = (unsigned*)(sclip + (size_t)N_EDGES * NH);  // [N, 8]
  float* ssum  = (float*)(smax + (size_t)N_NODES * NH);        // [N, 8]
  float* rowV  = ssum + (size_t)N_NODES * NH;                  // [N, 64]

  (void)in_sizes; (void)n_in; (void)out_size; (void)ws_size;

  k_init<<<(N_NODES * HD + 255) / 256, 256, 0, stream>>>(wV, ssum, rowV, smax);

  dim3 gq(N_NODES / 16, 12);
  k_qkv<<<gq, 32, 0, stream>>>(x, Wq, bq, Wk, bk, Wv, bv, Qh, Kh, Vh);

  k_edge<<<EDGE_WGS, 256, 0, stream>>>(edge_attr, We, be, src, dst,
                                       Qh, Kh, Aw, wE, sclip, smax);

  k_ssum<<<(N_EDGES * NH + 255) / 256, 256, 0, stream>>>(dst, sclip, smax, ssum);

  k_scatter<<<(N_EDGES * NH + 255) / 256, 256, 0, stream>>>(
      src, dst, sclip, smax, ssum, Vh, wE, wV, rowV);

  k_final<<<(N_NODES * HD + 255) / 256, 256, 0, stream>>>(rowV, VeRow, wV);
}